// GCNLayer_4810363372760
// MI455X (gfx1250) — compile-verified
//
#include <hip/hip_runtime.h>
#include <hip/hip_bf16.h>

typedef __attribute__((ext_vector_type(2))) float v2f;
typedef __attribute__((ext_vector_type(8))) float v8f;

#define NNODES 100000
#define DFEAT  128
#define BN_EPS 1e-5f

// ---------------- workspace layout (floats) ----------------
// [0,                N*128)              : agg
// [N*128,            N*128+N)            : deg_out -> norm_src (in place)
// [N*128+N,          N*128+2N)           : deg_in  -> norm_dst (in place)
// [N*128+2N,         +128)               : bn_sum
// ...+128                                : bn_sumsq
// ...+128                                : bn_scale
// ...+128                                : bn_shift

__global__ __launch_bounds__(256) void zero_kernel(float* __restrict__ p, size_t n) {
    size_t i = (size_t)blockIdx.x * blockDim.x + threadIdx.x;
    size_t stride = (size_t)gridDim.x * blockDim.x;
    for (; i < n; i += stride) p[i] = 0.0f;
}

__global__ __launch_bounds__(256) void degree_kernel(const int* __restrict__ src,
                                                     const int* __restrict__ dst,
                                                     float* __restrict__ dout,
                                                     float* __restrict__ din, int E) {
    int e = blockIdx.x * blockDim.x + threadIdx.x;
    if (e >= E) return;
    atomicAdd(&dout[src[e]], 1.0f);
    atomicAdd(&din[dst[e]], 1.0f);
}

__global__ __launch_bounds__(256) void norm_kernel(float* __restrict__ dout,
                                                   float* __restrict__ din, int n) {
    int i = blockIdx.x * blockDim.x + threadIdx.x;
    if (i >= n) return;
    dout[i] = rsqrtf(fmaxf(dout[i], 1.0f));
    din[i]  = rsqrtf(fmaxf(din[i], 1.0f));
}

// One wave32 per edge; lane l moves feats[src][4l..4l+3]*norm_src[src] into agg[dst].
__global__ __launch_bounds__(256) void scatter_kernel(const float* __restrict__ feats,
                                                      const int* __restrict__ src,
                                                      const int* __restrict__ dst,
                                                      const float* __restrict__ norm_src,
                                                      float* __restrict__ agg, int E) {
    long long t = (long long)blockIdx.x * blockDim.x + threadIdx.x;
    int e = (int)(t >> 5);
    if (e >= E) return;
    int lane = (int)(t & 31);
    int s = src[e], d = dst[e];
    float ns = norm_src[s];
    const float4* fr = (const float4*)(feats + (size_t)s * DFEAT);
    float4 v = fr[lane];                       // global_load_b128, L2-resident table
    float* ar = agg + (size_t)d * DFEAT + lane * 4;
    atomicAdd(ar + 0, v.x * ns);
    atomicAdd(ar + 1, v.y * ns);
    atomicAdd(ar + 2, v.z * ns);
    atomicAdd(ar + 3, v.w * ns);
}

// 8 waves per block; wave w owns 16x16 output tile (rows blockIdx.x*16, cols w*16).
// Dual f32 WMMA accumulators: conv = relu((agg*nd)@W + b), res = relu(feats@Wr + br),
// y = conv + res written to d_out; per-column BN partial sums accumulated atomically.
__global__ __launch_bounds__(256) void gemm_bn_kernel(
    const float* __restrict__ agg, const float* __restrict__ feats,
    const float* __restrict__ norm_dst,
    const float* __restrict__ W,  const float* __restrict__ b,
    const float* __restrict__ Wr, const float* __restrict__ br,
    float* __restrict__ y, float* __restrict__ bn_sum, float* __restrict__ bn_sumsq) {
    const int lane = threadIdx.x & 31;
    const int wave = threadIdx.x >> 5;
    const int rowBase = blockIdx.x * 16;
    const int n0 = wave * 16;
    const int m  = lane & 15;           // row (A) / col (B,C) within tile
    const int kh = (lane >> 4) * 2;     // K sub-offset for this half-wave: 0 or 2
    const int row = rowBase + m;
    const float nd = norm_dst[row];
    const float* aRow = agg   + (size_t)row * DFEAT;
    const float* fRow = feats + (size_t)row * DFEAT;

    v8f accC = {};
    v8f accR = {};
    #pragma unroll
    for (int k = 0; k < DFEAT; k += 4) {
        v2f a, f, bw, bwr;
        a.x = aRow[k + kh]     * nd;
        a.y = aRow[k + kh + 1] * nd;
        f.x = fRow[k + kh];
        f.y = fRow[k + kh + 1];
        bw.x  = W [(size_t)(k + kh)     * DFEAT + n0 + m];
        bw.y  = W [(size_t)(k + kh + 1) * DFEAT + n0 + m];
        bwr.x = Wr[(size_t)(k + kh)     * DFEAT + n0 + m];
        bwr.y = Wr[(size_t)(k + kh + 1) * DFEAT + n0 + m];
        accC = __builtin_amdgcn_wmma_f32_16x16x4_f32(false, a, false, bw,  (short)0, accC, false, false);
        accR = __builtin_amdgcn_wmma_f32_16x16x4_f32(false, f, false, bwr, (short)0, accR, false, false);
    }

    const int col  = n0 + m;
    const float bc  = b[col];
    const float brc = br[col];
    const int rOff = (lane >> 4) * 8;   // upper half-wave holds rows M+8
    float s8 = 0.0f, q8 = 0.0f;
    #pragma unroll
    for (int r = 0; r < 8; ++r) {
        float c = accC[r] + bc;  c = c > 0.0f ? c : 0.0f;
        float e = accR[r] + brc; e = e > 0.0f ? e : 0.0f;
        float v = c + e;
        y[(size_t)(rowBase + rOff + r) * DFEAT + col] = v;
        s8 += v;
        q8 += v * v;
    }
    // lanes l and l+16 hold the same column (rows 0-7 vs 8-15): fold, then 16 atomics/wave
    float s8o = __shfl_down(s8, 16);
    float q8o = __shfl_down(q8, 16);
    if (lane < 16) {
        atomicAdd(&bn_sum[col],   s8 + s8o);
        atomicAdd(&bn_sumsq[col], q8 + q8o);
    }
}

__global__ void bn_params_kernel(const float* __restrict__ sum, const float* __restrict__ sumsq,
                                 const float* __restrict__ gamma, const float* __restrict__ beta,
                                 float* __restrict__ scale, float* __restrict__ shift, int n) {
    int c = threadIdx.x;
    if (c >= DFEAT) return;
    float inv  = 1.0f / (float)n;
    float mean = sum[c] * inv;
    float var  = sumsq[c] * inv - mean * mean;   // biased variance, matches jnp.var
    float sc   = gamma[c] * rsqrtf(var + BN_EPS);
    scale[c] = sc;
    shift[c] = beta[c] - mean * sc;
}

__global__ __launch_bounds__(256) void bn_apply_kernel(float* __restrict__ y,
                                                       const float* __restrict__ scale,
                                                       const float* __restrict__ shift,
                                                       size_t nvec4) {
    size_t i = (size_t)blockIdx.x * blockDim.x + threadIdx.x;
    if (i >= nvec4) return;
    float4 v = ((float4*)y)[i];
    int col = (int)((i * 4) & (DFEAT - 1));
    v.x = v.x * scale[col + 0] + shift[col + 0];
    v.y = v.y * scale[col + 1] + shift[col + 1];
    v.z = v.z * scale[col + 2] + shift[col + 2];
    v.w = v.w * scale[col + 3] + shift[col + 3];
    ((float4*)y)[i] = v;
}

extern "C" void kernel_launch(void* const* d_in, const int* in_sizes, int n_in,
                              void* d_out, int out_size, void* d_ws, size_t ws_size,
                              hipStream_t stream) {
    const float* feats = (const float*)d_in[0];
    const int*   src   = (const int*)d_in[1];
    const int*   dst   = (const int*)d_in[2];
    const float* W     = (const float*)d_in[3];
    const float* b     = (const float*)d_in[4];
    const float* Wr    = (const float*)d_in[5];
    const float* br    = (const float*)d_in[6];
    const float* gamma = (const float*)d_in[7];
    const float* beta  = (const float*)d_in[8];
    float* out = (float*)d_out;

    const int N = NNODES;
    const int E = in_sizes[1];

    float* ws        = (float*)d_ws;
    float* agg       = ws;                                   // N*128
    float* deg_out   = agg + (size_t)N * DFEAT;              // N
    float* deg_in    = deg_out + N;                          // N
    float* bn_sum    = deg_in + N;                           // 128
    float* bn_sumsq  = bn_sum + DFEAT;                       // 128
    float* bn_scale  = bn_sumsq + DFEAT;                     // 128
    float* bn_shift  = bn_scale + DFEAT;                     // 128

    // 1) zero agg + degrees + bn accumulators
    size_t nz = (size_t)N * DFEAT + 2 * (size_t)N + 2 * DFEAT;
    zero_kernel<<<4096, 256, 0, stream>>>(ws, nz);

    // 2) degree counts
    degree_kernel<<<(E + 255) / 256, 256, 0, stream>>>(src, dst, deg_out, deg_in, E);

    // 3) norms (in place on degree buffers)
    norm_kernel<<<(N + 255) / 256, 256, 0, stream>>>(deg_out, deg_in, N);

    // 4) edge scatter: one wave32 per edge
    long long sthreads = (long long)E * 32;
    int sblocks = (int)((sthreads + 255) / 256);
    scatter_kernel<<<sblocks, 256, 0, stream>>>(feats, src, dst, deg_out, agg, E);

    // 5) dual WMMA GEMM + bias + relu + sum + BN partial stats  (N%16==0 -> 6250 tiles)
    gemm_bn_kernel<<<N / 16, 256, 0, stream>>>(agg, feats, deg_in, W, b, Wr, br,
                                               out, bn_sum, bn_sumsq);

    // 6) BN affine params
    bn_params_kernel<<<1, DFEAT, 0, stream>>>(bn_sum, bn_sumsq, gamma, beta,
                                              bn_scale, bn_shift, N);

    // 7) apply BN in place on d_out (float4 vectorized)
    size_t nvec4 = (size_t)N * DFEAT / 4;
    bn_apply_kernel<<<(int)((nvec4 + 255) / 256), 256, 0, stream>>>(out, bn_scale, bn_shift, nvec4);
}